// SAGEStack_20143396618810
// MI455X (gfx1250) — compile-verified
//
#include <hip/hip_runtime.h>

typedef float v2f __attribute__((ext_vector_type(2)));
typedef float v8f __attribute__((ext_vector_type(8)));

#define NN    50000
#define NE    640000
#define HID   128
#define LDSW  132          // padded LDS row stride (floats): banks = 4*l + k, conflict-free
#define NLAY  3
#define LN_EPS 1e-5f

__device__ __forceinline__ v8f wmma_f32_k4(v2f a, v2f b, v8f c) {
    // D = A(16x4,f32) * B(4x16,f32) + C(16x16,f32)
    return __builtin_amdgcn_wmma_f32_16x16x4_f32(false, a, false, b, (short)0, c, false, false);
}

// ---------------------------------------------------------------------------
// deg[d] += 1 for every edge (d = edge_index[1][e])
// ---------------------------------------------------------------------------
__global__ __launch_bounds__(256) void degree_kernel(const int* __restrict__ ei,
                                                     float* __restrict__ deg) {
    int i = blockIdx.x * blockDim.x + threadIdx.x;
    if (i < NE) atomicAdd(&deg[ei[NE + i]], 1.0f);
}

// ---------------------------------------------------------------------------
// agg[dst] += h[src]  (one wave per edge, 4 floats per lane)
// ---------------------------------------------------------------------------
__global__ __launch_bounds__(256) void scatter_kernel(const int* __restrict__ ei,
                                                      const float* __restrict__ h,
                                                      float* __restrict__ agg) {
    int wid = blockIdx.x * (blockDim.x >> 5) + (threadIdx.x >> 5);
    if (wid >= NE) return;
    int lane = threadIdx.x & 31;
    int s = ei[wid];
    int d = ei[NE + wid];
    const float4 v = *(const float4*)(h + (size_t)s * HID + lane * 4);
    float* dst = agg + (size_t)d * HID + lane * 4;
    atomicAdd(dst + 0, v.x);
    atomicAdd(dst + 1, v.y);
    atomicAdd(dst + 2, v.z);
    atomicAdd(dst + 3, v.w);
}

// ---------------------------------------------------------------------------
// h = x @ W + b   (16 rows per block, 8 waves = 8 col tiles of 16, K=128)
// ---------------------------------------------------------------------------
__global__ __launch_bounds__(256) void in_gemm_kernel(const float* __restrict__ x,
                                                      const float* __restrict__ W,
                                                      const float* __restrict__ b,
                                                      float* __restrict__ h) {
    __shared__ float sX[16 * LDSW];
    const int row0 = blockIdx.x * 16;
    const int t = threadIdx.x;

    // stage 16x128 A-tile into LDS (float4, coalesced)
    #pragma unroll
    for (int i = 0; i < 2; ++i) {
        int f  = t + i * 256;          // float4 index within tile [0,512)
        int r  = f >> 5;               // row 0..15
        int c4 = f & 31;               // float4 col 0..31
        float4 v = ((const float4*)(x + (size_t)(row0 + r) * HID))[c4];
        *(float4*)&sX[r * LDSW + c4 * 4] = v;
    }
    __syncthreads();

    const int wave = t >> 5, lane = t & 31;
    const int n0 = wave * 16;
    const int m = lane & 15, hi = lane >> 4;

    v8f c = {};
    for (int k0 = 0; k0 < HID; k0 += 4) {
        v2f a = *(const v2f*)&sX[m * LDSW + k0 + 2 * hi];
        v2f bb;
        bb.x = W[(k0 + 2 * hi) * HID + n0 + m];
        bb.y = W[(k0 + 2 * hi + 1) * HID + n0 + m];
        c = wmma_f32_k4(a, bb, c);
    }

    const float bias = b[n0 + m];
    float* out = h + (size_t)row0 * HID;
    #pragma unroll
    for (int v = 0; v < 8; ++v) {
        int M = v + 8 * hi;
        out[(size_t)M * HID + n0 + m] = c[v] + bias;
    }
}

// ---------------------------------------------------------------------------
// h_out = LayerNorm( relu( (agg/deg)@Wl + bl + h_in@Wr ) + h_in )
// ---------------------------------------------------------------------------
__global__ __launch_bounds__(256) void sage_layer_kernel(const float* __restrict__ agg,
                                                         const float* __restrict__ deg,
                                                         const float* __restrict__ h_in,
                                                         const float* __restrict__ Wl,
                                                         const float* __restrict__ bl,
                                                         const float* __restrict__ Wr,
                                                         const float* __restrict__ ln_g,
                                                         const float* __restrict__ ln_b,
                                                         float* __restrict__ h_out) {
    __shared__ float sA[16 * LDSW];   // agg * inv_deg tile, later reused for LN staging
    __shared__ float sH[16 * LDSW];   // h_in tile
    const int row0 = blockIdx.x * 16;
    const int t = threadIdx.x;

    #pragma unroll
    for (int i = 0; i < 2; ++i) {
        int f  = t + i * 256;
        int r  = f >> 5;
        int c4 = f & 31;
        float invd = 1.0f / fmaxf(deg[row0 + r], 1.0f);
        float4 va = ((const float4*)(agg + (size_t)(row0 + r) * HID))[c4];
        va.x *= invd; va.y *= invd; va.z *= invd; va.w *= invd;
        *(float4*)&sA[r * LDSW + c4 * 4] = va;
        float4 vh = ((const float4*)(h_in + (size_t)(row0 + r) * HID))[c4];
        *(float4*)&sH[r * LDSW + c4 * 4] = vh;
    }
    __syncthreads();

    const int wave = t >> 5, lane = t & 31;
    const int n0 = wave * 16;
    const int m = lane & 15, hi = lane >> 4;

    v8f c = {};
    for (int k0 = 0; k0 < HID; k0 += 4) {          // (agg/deg) @ Wl
        v2f a = *(const v2f*)&sA[m * LDSW + k0 + 2 * hi];
        v2f bb;
        bb.x = Wl[(k0 + 2 * hi) * HID + n0 + m];
        bb.y = Wl[(k0 + 2 * hi + 1) * HID + n0 + m];
        c = wmma_f32_k4(a, bb, c);
    }
    for (int k0 = 0; k0 < HID; k0 += 4) {          // + h_in @ Wr
        v2f a = *(const v2f*)&sH[m * LDSW + k0 + 2 * hi];
        v2f bb;
        bb.x = Wr[(k0 + 2 * hi) * HID + n0 + m];
        bb.y = Wr[(k0 + 2 * hi + 1) * HID + n0 + m];
        c = wmma_f32_k4(a, bb, c);
    }

    const float bias = bl[n0 + m];
    __syncthreads();                                // all waves done reading sA
    #pragma unroll
    for (int v = 0; v < 8; ++v) {                   // bias + relu + residual -> LN stage
        int M = v + 8 * hi;
        float val = fmaxf(c[v] + bias, 0.0f) + sH[M * LDSW + n0 + m];
        sA[M * LDSW + n0 + m] = val;
    }
    __syncthreads();

    // LayerNorm: 16 threads per row, 8 elems per thread, shfl_xor tree in groups of 16
    const int r   = t >> 4;     // row 0..15
    const int l16 = t & 15;
    float s1 = 0.0f, s2 = 0.0f;
    float vals[8];
    #pragma unroll
    for (int j = 0; j < 8; ++j) {
        float xv = sA[r * LDSW + l16 + 16 * j];
        vals[j] = xv;
        s1 += xv;
        s2 += xv * xv;
    }
    #pragma unroll
    for (int off = 8; off >= 1; off >>= 1) {
        s1 += __shfl_xor(s1, off, 16);
        s2 += __shfl_xor(s2, off, 16);
    }
    const float mean = s1 * (1.0f / HID);
    const float var  = s2 * (1.0f / HID) - mean * mean;
    const float rstd = rsqrtf(var + LN_EPS);

    float* out = h_out + (size_t)(row0 + r) * HID;
    #pragma unroll
    for (int j = 0; j < 8; ++j) {
        int col = l16 + 16 * j;
        out[col] = (vals[j] - mean) * rstd * ln_g[col] + ln_b[col];
    }
}

// ---------------------------------------------------------------------------
extern "C" void kernel_launch(void* const* d_in, const int* in_sizes, int n_in,
                              void* d_out, int out_size, void* d_ws, size_t ws_size,
                              hipStream_t stream) {
    const float* x   = (const float*)d_in[0];
    const int*   ei  = (const int*)d_in[1];
    const float* inW = (const float*)d_in[2];
    const float* inb = (const float*)d_in[3];
    const float* Wl  = (const float*)d_in[4];
    const float* bl  = (const float*)d_in[5];
    const float* Wr  = (const float*)d_in[6];
    const float* lng = (const float*)d_in[7];
    const float* lnb = (const float*)d_in[8];
    float* out = (float*)d_out;

    // workspace layout
    char* ws = (char*)d_ws;
    const size_t degBytes = ((size_t)NN * 4 + 511) & ~(size_t)511;
    const size_t hBytes   = (size_t)NN * HID * 4;
    float* deg = (float*)ws;
    float* agg = (float*)(ws + degBytes);
    float* hA  = (float*)(ws + degBytes + hBytes);
    float* hB  = (float*)(ws + degBytes + 2 * hBytes);

    // degrees (once)
    hipMemsetAsync(deg, 0, (size_t)NN * 4, stream);
    degree_kernel<<<(NE + 255) / 256, 256, 0, stream>>>(ei, deg);

    // input projection
    in_gemm_kernel<<<NN / 16, 256, 0, stream>>>(x, inW, inb, hA);

    float* cur = hA;
    float* nxt = hB;
    for (int i = 0; i < NLAY; ++i) {
        hipMemsetAsync(agg, 0, hBytes, stream);
        scatter_kernel<<<NE / 8, 256, 0, stream>>>(ei, cur, agg);
        float* dst = (i == NLAY - 1) ? out : nxt;
        sage_layer_kernel<<<NN / 16, 256, 0, stream>>>(
            agg, deg, cur,
            Wl + (size_t)i * HID * HID, bl + (size_t)i * HID,
            Wr + (size_t)i * HID * HID,
            lng + (size_t)i * HID, lnb + (size_t)i * HID, dst);
        float* tmp = cur; cur = dst; nxt = tmp;
    }
}